// SAGEConv_1872605741716
// MI455X (gfx1250) — compile-verified
//
#include <hip/hip_runtime.h>

// ---------------------------------------------------------------------------
// SAGEConv ('pool' aggregator) for MI455X / gfx1250.
//   h_pool = relu(X @ Wp^T + bp)                    [65536 x 64]   fp32 WMMA
//   neigh[d] = max over edges (s->d) of h_pool[s]   GLOBAL_ATOMIC_MAX_NUM_F32
//   out = X @ Ws^T + bs + neigh @ Wn^T + bn         [65536 x 128]  fp32 WMMA
// GEMMs use V_WMMA_F32_16X16X4_F32 (exact fp32, matches reference math).
// Scatter-max uses the native non-returning f32 max atomic at device scope
// (h_pool and neigh are both L2-resident: 16 MB each vs 192 MB L2).
// ---------------------------------------------------------------------------

#define N_NODES   65536
#define IN_FEATS  64
#define OUT_FEATS 128
#define N_EDGES   1048576

typedef __attribute__((ext_vector_type(2))) float v2f;
typedef __attribute__((ext_vector_type(8))) float v8f;

__device__ __forceinline__ v8f wmma_f32_16x16x4(v2f a, v2f b, v8f c) {
  // 8 args: (neg_a, A, neg_b, B, c_mod, C, reuse_a, reuse_b)
  return __builtin_amdgcn_wmma_f32_16x16x4_f32(
      false, a, false, b, (short)0, c, false, false);
}

// Native CDNA5 f32 max atomic, non-returning (STOREcnt), device scope.
// IEEE maximumNumber semantics == jnp maximum-based segment_max for our data.
__device__ __forceinline__ void atomic_max_f32_dev(float* p, float v) {
  asm volatile("global_atomic_max_num_f32 %0, %1, off scope:SCOPE_DEV"
               :
               : "v"(p), "v"(v)
               : "memory");
}

// -------------------------------------------------------------- zero scratch
__global__ void sage_zero_kernel(float* __restrict__ p, int n) {
  int i = blockIdx.x * blockDim.x + threadIdx.x;
  if (i < n) p[i] = 0.0f;
}

// -------------------------------------------- h_pool = relu(X @ Wp^T + bp)
// grid.x = N_NODES/16 row tiles; 4 waves/block, wave w owns cols [16w,16w+16)
__global__ void sage_hpool_kernel(const float* __restrict__ feat,
                                  const float* __restrict__ wp,
                                  const float* __restrict__ bp,
                                  float* __restrict__ hpool) {
  const int lane = threadIdx.x & 31;
  const int wave = threadIdx.x >> 5;      // 0..3
  const int m0   = blockIdx.x * 16;
  const int n0   = wave * 16;
  const int r    = lane & 15;
  const int half = lane >> 4;             // 0: K pair (0,1), 1: K pair (2,3)

  const float* arow = feat + (size_t)(m0 + r) * IN_FEATS;  // A row (M = m0+r)
  const float* brow = wp   + (size_t)(n0 + r) * IN_FEATS;  // B col n0+r == Wp row

  v8f acc = {};
#pragma unroll
  for (int k0 = 0; k0 < IN_FEATS; k0 += 4) {
    v2f a = *(const v2f*)(arow + k0 + 2 * half);
    v2f b = *(const v2f*)(brow + k0 + 2 * half);
    acc = wmma_f32_16x16x4(a, b, acc);
  }

  const float bias = bp[n0 + r];
#pragma unroll
  for (int j = 0; j < 8; ++j) {
    float v = acc[j] + bias;
    v = v > 0.0f ? v : 0.0f;              // relu
    hpool[(size_t)(m0 + 8 * half + j) * IN_FEATS + (n0 + r)] = v;
  }
}

// ---------------------------- neigh[dst] = max(neigh[dst], h_pool[src]) -----
// One thread per (edge, 4-float chunk): 16 threads/edge, float4 gather from
// L2-resident h_pool, four native f32 max atomics into L2-resident neigh.
__global__ void sage_edge_max_kernel(const int* __restrict__ ei,
                                     const float* __restrict__ hpool,
                                     float* __restrict__ neigh) {
  const unsigned tid = blockIdx.x * blockDim.x + threadIdx.x;
  const unsigned e = tid >> 4;
  const unsigned c = (tid & 15u) << 2;
  if (e >= N_EDGES) return;

  const int src = ei[e];                  // edge_index[0][e]
  const int dst = ei[N_EDGES + e];        // edge_index[1][e]

  const float4 v = *(const float4*)(hpool + (size_t)src * IN_FEATS + c);
  float* d = neigh + (size_t)dst * IN_FEATS + c;
  atomic_max_f32_dev(d + 0, v.x);
  atomic_max_f32_dev(d + 1, v.y);
  atomic_max_f32_dev(d + 2, v.z);
  atomic_max_f32_dev(d + 3, v.w);
}

// ------------- out = X @ Ws^T + bs + neigh @ Wn^T + bn ----------------------
// grid.x = N_NODES/16 row tiles; 8 waves/block, wave w owns cols [16w,16w+16)
__global__ void sage_out_kernel(const float* __restrict__ feat,
                                const float* __restrict__ neigh,
                                const float* __restrict__ wsl,
                                const float* __restrict__ bsl,
                                const float* __restrict__ wnl,
                                const float* __restrict__ bnl,
                                float* __restrict__ out) {
  const int lane = threadIdx.x & 31;
  const int wave = threadIdx.x >> 5;      // 0..7
  const int m0   = blockIdx.x * 16;
  const int n0   = wave * 16;
  const int r    = lane & 15;
  const int half = lane >> 4;

  v8f acc = {};

  // X @ Ws^T contribution
  {
    const float* arow = feat + (size_t)(m0 + r) * IN_FEATS;
    const float* brow = wsl  + (size_t)(n0 + r) * IN_FEATS;
#pragma unroll
    for (int k0 = 0; k0 < IN_FEATS; k0 += 4) {
      v2f a = *(const v2f*)(arow + k0 + 2 * half);
      v2f b = *(const v2f*)(brow + k0 + 2 * half);
      acc = wmma_f32_16x16x4(a, b, acc);
    }
  }
  // neigh @ Wn^T contribution (accumulates into same C)
  {
    const float* arow = neigh + (size_t)(m0 + r) * IN_FEATS;
    const float* brow = wnl   + (size_t)(n0 + r) * IN_FEATS;
#pragma unroll
    for (int k0 = 0; k0 < IN_FEATS; k0 += 4) {
      v2f a = *(const v2f*)(arow + k0 + 2 * half);
      v2f b = *(const v2f*)(brow + k0 + 2 * half);
      acc = wmma_f32_16x16x4(a, b, acc);
    }
  }

  const float bias = bsl[n0 + r] + bnl[n0 + r];
#pragma unroll
  for (int j = 0; j < 8; ++j) {
    out[(size_t)(m0 + 8 * half + j) * OUT_FEATS + (n0 + r)] = acc[j] + bias;
  }
}

// ---------------------------------------------------------------------------
extern "C" void kernel_launch(void* const* d_in, const int* in_sizes, int n_in,
                              void* d_out, int out_size, void* d_ws, size_t ws_size,
                              hipStream_t stream) {
  const float* feat = (const float*)d_in[0];   // [65536, 64]
  const float* wp   = (const float*)d_in[1];   // [64, 64]
  const float* bp   = (const float*)d_in[2];   // [64]
  const float* wsl  = (const float*)d_in[3];   // [128, 64]
  const float* bsl  = (const float*)d_in[4];   // [128]
  const float* wnl  = (const float*)d_in[5];   // [128, 64]
  const float* bnl  = (const float*)d_in[6];   // [128]
  const int*   ei   = (const int*)d_in[7];     // [2, 1048576]
  float* out = (float*)d_out;                  // [65536, 128]

  // Scratch layout: [0,16MB) h_pool ; [16MB,32MB) neigh
  float* hpool = (float*)d_ws;
  float* neigh = hpool + (size_t)N_NODES * IN_FEATS;

  const int nfeat = N_NODES * IN_FEATS;        // 4M floats each

  // 1) neigh = 0 (relu output >= 0, so max-with-0 == segment_max w/ 0 fallback)
  sage_zero_kernel<<<nfeat / 256, 256, 0, stream>>>(neigh, nfeat);

  // 2) h_pool GEMM (WMMA f32): 4096 row tiles x 4 col-tile waves
  sage_hpool_kernel<<<N_NODES / 16, 128, 0, stream>>>(feat, wp, bp, hpool);

  // 3) edge gather + scatter atomic f32 max: 16 threads/edge
  sage_edge_max_kernel<<<(N_EDGES * 16) / 256, 256, 0, stream>>>(ei, hpool, neigh);

  // 4) output GEMMs (WMMA f32): 4096 row tiles x 8 col-tile waves
  sage_out_kernel<<<N_NODES / 16, 256, 0, stream>>>(feat, neigh, wsl, bsl,
                                                    wnl, bnl, out);
}